// ExpSynTorch_46024869544113
// MI455X (gfx1250) — compile-verified
//
#include <hip/hip_runtime.h>
#include <hip/hip_bf16.h>

// ExpSyn diagonal linear recurrence: isyn[b,t,n] = beta[n]*isyn[b,t-1,n] + x[b,t,n]
// B=16, T=4096, N=512, fp32. HBM-bound (268 MB total -> ~11.5us @ 23.3 TB/s).
// Strategy: per-(b, 32-neuron tile) workgroup, 8 waves; T walked in segments of 512
// double-buffered in LDS via CDNA5 async global->LDS loads (non-temporal, data is
// consumed exactly once); intra-segment scan parallelized across waves with an LDS
// carry-prefix fixup (exact beta^64 jump factor). Output uses non-temporal stores
// (write-once stream, 268 MB footprint > 192 MB L2, so don't pollute L2).

#define ES_B    16
#define ES_T    4096
#define ES_N    512
#define ES_NT   32              // neurons per workgroup (one per lane)
#define ES_WAVES 8
#define ES_TSEG 512             // timesteps per LDS segment
#define ES_LC   (ES_TSEG / ES_WAVES)   // 64 timesteps per wave sub-chunk
#define ES_NSEG (ES_T / ES_TSEG)       // 8 segments
#define ES_DT   1e-4f

// Issue 16 async b128 loads covering this wave's sub-chunk (64 t x 32 n x 4B = 8KB).
// Lane l covers t-offset (l>>3) and 4 consecutive neurons starting at (l&7)*4.
__device__ __forceinline__ void es_prefetch_subchunk(const float* gbase,
                                                     unsigned lds_byte_base) {
#pragma unroll
    for (int i = 0; i < ES_LC / 4; ++i) {
        unsigned long long ga = (unsigned long long)(gbase + (size_t)(4 * i) * ES_N);
        unsigned lb = lds_byte_base + (unsigned)(4 * i) * (ES_NT * 4u);
        asm volatile("global_load_async_to_lds_b128 %0, %1, off th:TH_LOAD_NT"
                     :: "v"(lb), "v"(ga) : "memory");
    }
}

__global__ __launch_bounds__(256, 2) void ExpSyn_scan_kernel(
    const float* __restrict__ data,     // (B, T, N)
    const float* __restrict__ tau_syn,  // (1, N)
    float* __restrict__ out)            // (B, T, N)
{
    extern __shared__ float smem[];     // [2][TSEG][NT] segment bufs + [WAVES][NT] finals
    float* bufs   = smem;
    float* finals = smem + 2 * ES_TSEG * ES_NT;

    const int tid  = threadIdx.x;
    const int lane = tid & 31;          // neuron within tile
    const int wid  = tid >> 5;          // wave id = sub-chunk id
    const int wg   = blockIdx.x;        // 0..255
    const int b    = wg >> 4;           // B = 16
    const int n0   = (wg & 15) * ES_NT; // neuron tile base

    const float tau    = tau_syn[n0 + lane];
    const float beta   = __expf(-ES_DT / tau);
    const float betaLc = __expf(-(ES_DT * (float)ES_LC) / tau); // beta^64, exact

    const size_t baseBT = (size_t)b * ES_T * ES_N + (size_t)n0;
    const int tsub = wid * ES_LC;       // sub-chunk start within segment
    const int l_t  = lane >> 3;         // 0..3   (t offset for async b128)
    const int l_n  = (lane & 7) * 4;    // 0..28  (n offset for async b128)

    float seg_carry = 0.0f;             // isyn state entering current segment

    // -------- prologue: prefetch segment 0 into buffer 0 --------
    {
        const float* g0 = data + baseBT + (size_t)(tsub + l_t) * ES_N + l_n;
        unsigned lb0 = ((unsigned)(unsigned long long)bufs)
                     + (unsigned)((tsub + l_t) * ES_NT + l_n) * 4u;
        es_prefetch_subchunk(g0, lb0);
    }

    for (int s = 0; s < ES_NSEG; ++s) {
        const int t0  = s * ES_TSEG;
        const int cur = s & 1;
        float* buf = bufs + cur * (ES_TSEG * ES_NT);

        // -------- prefetch next segment into the other buffer --------
        if (s + 1 < ES_NSEG) {
            float* nbuf = bufs + ((s + 1) & 1) * (ES_TSEG * ES_NT);
            const float* g0 = data + baseBT
                            + (size_t)(t0 + ES_TSEG + tsub + l_t) * ES_N + l_n;
            unsigned lb0 = ((unsigned)(unsigned long long)nbuf)
                         + (unsigned)((tsub + l_t) * ES_NT + l_n) * 4u;
            asm volatile("s_wait_dscnt 0x0" ::: "memory"); // LDS reads of reused buf done
            es_prefetch_subchunk(g0, lb0);
            asm volatile("s_wait_asynccnt 0x10" ::: "memory"); // current seg (16 older) done
        } else {
            asm volatile("s_wait_asynccnt 0x0" ::: "memory");
        }

        // -------- phase 1: local scan of own sub-chunk (zero init), in place --------
        float acc = 0.0f;
        float* p = buf + tsub * ES_NT + lane;
#pragma unroll 8
        for (int tl = 0; tl < ES_LC; ++tl) {
            float x = p[tl * ES_NT];
            acc = fmaf(beta, acc, x);
            p[tl * ES_NT] = acc;        // local_t (carry-free)
        }
        finals[wid * ES_NT + lane] = acc;
        __syncthreads();

        // -------- phase 2: redundant per-thread carry prefix over sub-chunks --------
        float cw = seg_carry;           // carry entering THIS wave's sub-chunk
        float cn = seg_carry;           // carry entering NEXT segment
#pragma unroll
        for (int k = 0; k < ES_WAVES; ++k) {
            float fk = finals[k * ES_NT + lane];
            if (k < wid) cw = fmaf(betaLc, cw, fk);
            cn = fmaf(betaLc, cn, fk);
        }
        __syncthreads();                // finals may be overwritten next segment
        seg_carry = cn;

        // -------- phase 3: apply carry, stream non-temporal stores to global --------
        float pw = beta;                // beta^(tl+1)
        float* op = out + baseBT + (size_t)(t0 + tsub) * ES_N + lane;
#pragma unroll 8
        for (int tl = 0; tl < ES_LC; ++tl) {
            float v = fmaf(pw, cw, p[tl * ES_NT]);  // y = local + beta^{t+1}*carry
            __builtin_nontemporal_store(v, op + (size_t)tl * ES_N);
            pw *= beta;
        }
    }
}

extern "C" void kernel_launch(void* const* d_in, const int* in_sizes, int n_in,
                              void* d_out, int out_size, void* d_ws, size_t ws_size,
                              hipStream_t stream) {
    const float* data = (const float*)d_in[0];  // (16, 4096, 512) fp32
    const float* tau  = (const float*)d_in[1];  // (1, 512) fp32
    float* out = (float*)d_out;                 // (16, 4096, 512) fp32

    const size_t shmem = (size_t)(2 * ES_TSEG * ES_NT + ES_WAVES * ES_NT) * sizeof(float);
    const dim3 grid(ES_B * (ES_N / ES_NT));     // 256 workgroups
    const dim3 block(ES_WAVES * 32);            // 256 threads = 8 wave32

    ExpSyn_scan_kernel<<<grid, block, shmem, stream>>>(data, tau, out);
}